// SFNOBlock3d_31456340476382
// MI455X (gfx1250) — compile-verified
//
#include <hip/hip_runtime.h>
#include <math.h>

#ifndef M_PI
#define M_PI 3.14159265358979323846
#endif

typedef __attribute__((ext_vector_type(2))) float v2f;
typedef __attribute__((ext_vector_type(8))) float v8f;

// Problem constants
#define BSZ   64
#define CIN   8
#define COUT  8
#define NLAT  128
#define NLON  256
#define MM0   16
#define MM1   16
#define RROWS (BSZ*CIN*NLAT)      // 65536 rows (b,c,k)
#define INFLT (2*CIN*MM0*MM1)     // 4096
#define HID   (4*COUT*MM0*MM1)    // 8192

__device__ __forceinline__ float gelu_exact(float v) {
    return 0.5f * v * (1.0f + erff(v * 0.70710678118654752f));
}

// ---------------------------------------------------------------------------
// Init: Clenshaw-Curtis weights + ortho-normalized associated Legendre
// recurrence (f64):
//   Wfwd2[j=2m+ri][k][l] = P_l^m(ct_k) * CSphase * wq_k    (forward B)
//   Pinv2[j=2m+ri][l][k] = P_l^m(ct_k) * CSphase           (inverse B)
// ---------------------------------------------------------------------------
__global__ __launch_bounds__(128) void init_legendre_k(float* __restrict__ Wfwd2,
                                                       float* __restrict__ Pinv2) {
    const int k = threadIdx.x;                 // quadrature node 0..127
    const double N = 127.0;
    const double tj = M_PI * (double)k / N;
    double w = 2.0 / N;
    for (int j = 1; j <= 63; ++j)
        w -= 2.0 * cos(2.0 * j * tj) * (2.0 / N) / (4.0 * (double)j * (double)j - 1.0);
    if (k == 0 || k == 127) w *= 0.5;
    const double x = cos(M_PI * (double)(127 - k) / N);   // ct[k] = reversed nodes

    double vdm[16][16];
    for (int m = 0; m < 16; ++m)
        for (int l = 0; l < 16; ++l) vdm[m][l] = 0.0;
    vdm[0][0] = 1.0 / sqrt(4.0 * M_PI);
    for (int l = 1; l < 16; ++l) {
        vdm[l-1][l] = sqrt(2.0*l + 1.0) * x * vdm[l-1][l-1];
        vdm[l][l]   = sqrt((2.0*l + 1.0) * (1.0 + x) * (1.0 - x) / (2.0 * l)) * vdm[l-1][l-1];
    }
    for (int l = 2; l < 16; ++l) {
        for (int m = 0; m <= l - 2; ++m) {
            double a = sqrt((2.0*l - 1.0)/(double)(l - m) * (2.0*l + 1.0)/(double)(l + m));
            double b = sqrt((l + m - 1.0)/(double)(l - m) * (2.0*l + 1.0)/(2.0*l - 3.0)
                            * (l - m - 1.0)/(double)(l + m));
            vdm[m][l] = a * x * vdm[m][l-1] - b * vdm[m][l-2];
        }
    }
    for (int m = 0; m < 16; ++m) {
        const double sgn = (m & 1) ? -1.0 : 1.0;          // vdm[1::2] *= -1
        for (int l = 0; l < 16; ++l) {
            const double v = vdm[m][l] * sgn;
            const float pv = (float)v;
            const float wv = (float)(v * w);
            Wfwd2[((2*m)   * NLAT + k) * 16 + l] = wv;
            Wfwd2[((2*m+1) * NLAT + k) * 16 + l] = wv;
            Pinv2[((2*m)   * 16 + l) * NLAT + k] = pv;
            Pinv2[((2*m+1) * 16 + l) * NLAT + k] = pv;
        }
    }
}

// DFT bases. Forward (rfft, m<16, scaled 2pi/NLON): basisF[w][j=2m+ri].
// Inverse (irfft * NLON, m<16, Im(X0) ignored): basisI[j][w].
__global__ __launch_bounds__(256) void init_basis_k(float* __restrict__ basisF,
                                                    float* __restrict__ basisI) {
    const int t = blockIdx.x * 256 + threadIdx.x;   // 8192 = 256*32
    const int w = t >> 5, j = t & 31, m = j >> 1, ri = j & 1;
    const double th = 2.0 * M_PI * (double)m * (double)w / 256.0;
    const double s  = 2.0 * M_PI / 256.0;
    basisF[w * 32 + j] = (float)(ri ? -sin(th) * s : cos(th) * s);
    float g;
    if (m == 0) g = ri ? 0.0f : 1.0f;
    else        g = (float)(ri ? -2.0 * sin(th) : 2.0 * cos(th));
    basisI[j * 256 + w] = g;
}

// 1x1 conv bypass (float4 I/O): byp[b][o][k][w] = cb[o] + sum_ci cw[o][ci]*x[..]
__global__ __launch_bounds__(256) void bypass_k(const float* __restrict__ x,
                                                const float* __restrict__ cw,
                                                const float* __restrict__ cb,
                                                float* __restrict__ byp) {
    const int t  = blockIdx.x * 256 + threadIdx.x;   // 524288 = 64*32768/4
    const int b  = t >> 13;                          // / 8192 quads
    const int sp = (t & 8191) * 4;                   // quad offset in k*256+w
    const float4* xb = (const float4*)(x + b * (CIN * 32768) + sp);
    float4 xv[CIN];
#pragma unroll
    for (int ci = 0; ci < CIN; ++ci) xv[ci] = xb[ci * 8192];
#pragma unroll
    for (int o = 0; o < COUT; ++o) {
        float4 acc; acc.x = acc.y = acc.z = acc.w = cb[o];
#pragma unroll
        for (int ci = 0; ci < CIN; ++ci) {
            const float wv = cw[o * CIN + ci];
            acc.x += wv * xv[ci].x; acc.y += wv * xv[ci].y;
            acc.z += wv * xv[ci].z; acc.w += wv * xv[ci].w;
        }
        *(float4*)(byp + b * (COUT * 32768) + o * 32768 + sp) = acc;
    }
}

// ---------------------------------------------------------------------------
// Generic f32 WMMA GEMM: C[m,n] = act( sum_k A[m,k]*B[k,n] + bias[n] + D[m,n] )
// Compile-time load modes (no branches in the hot loop):
//   AMODE 0: A k-contiguous (a_cs==1), b128 loads     1: strided scalar gather
//   BMODE 0: B n-contiguous (b_cs==1), b128 loads     1: B k-contiguous (W^T)
// One 16x16 tile per wave, wave-private LDS slice (intra-wave LDS is in-order;
// wave_barrier pins compiler ordering — no s_barrier needed). Latency hidden
// by occupancy + global_prefetch of the next k-chunk (speculative, safe OOB).
// ---------------------------------------------------------------------------
template<int AMODE, int BMODE>
__global__ __launch_bounds__(256) void gemm_f32_wmma(
    const float* __restrict__ A, const float* __restrict__ Bm,
    const float* __restrict__ bias, const float* __restrict__ D,
    float* __restrict__ C,
    int M, int N, int K, int act,
    int a_rs, int a_cs, int a_bs,
    int b_rs, int b_cs, int b_bs,
    int c_rs, int c_cs, int c_bs,
    int d_rs, int d_cs, int d_bs)
{
    __shared__ float As[8][16][20];   // pitch 20: b128-aligned rows, 20i%64 distinct
    __shared__ float Bs[8][16][20];

    const int lane = threadIdx.x & 31;
    const int wave = threadIdx.x >> 5;
    const int half = lane >> 4;
    const int l16  = lane & 15;
    const int q    = lane >> 2;        // 0..7 (vector-load row)
    const int r4   = (lane & 3) * 4;   // 0,4,8,12 (vector-load col quad)

    const int ntiles = N >> 4;
    const int tiles  = (M >> 4) * ntiles;
    const int tid = blockIdx.x * 8 + wave;
    if (tid >= tiles) return;          // wave-uniform: EXEC stays all-ones
    const int tm = tid / ntiles;
    const int tn = tid - tm * ntiles;

    const float* Abase = A  + blockIdx.z * a_bs + tm * 16 * a_rs;
    const float* Bbase = Bm + blockIdx.z * b_bs + tn * 16 * b_cs;

    const float* aP0;  const float* aP1 = Abase;  int aStep;
    int aOff[8];
    if constexpr (AMODE == 0) {
        aP0 = Abase +  q      * a_rs + r4;
        aP1 = Abase + (q + 8) * a_rs + r4;
        aStep = 16;
    } else {
        aP0 = Abase + l16 * a_cs;
        aStep = 16 * a_cs;
#pragma unroll
        for (int p = 0; p < 8; ++p) aOff[p] = (p * 2 + half) * a_rs;
    }
    const float* bP0;  const float* bP1;  int bStep;
    if constexpr (BMODE == 0) {        // rows are k, n contiguous
        bP0 = Bbase +  q      * b_rs + r4;
        bP1 = Bbase + (q + 8) * b_rs + r4;
        bStep = 16 * b_rs;
    } else {                           // rows are n, k contiguous (transposed W)
        bP0 = Bbase +  q      * b_cs + r4;
        bP1 = Bbase + (q + 8) * b_cs + r4;
        bStep = 16;
    }

    v8f acc = {0.f, 0.f, 0.f, 0.f, 0.f, 0.f, 0.f, 0.f};
    const int nc = K >> 4;
    for (int c = 0; c < nc; ++c) {
        // ---- stage this k-chunk: global -> regs -> this wave's LDS slice ----
        float4 fa0, fa1, fb0, fb1;
        float  fsa[8];
        if constexpr (AMODE == 0) {
            fa0 = *(const float4*)aP0;  fa1 = *(const float4*)aP1;
            aP0 += aStep;  aP1 += aStep;
            __builtin_prefetch(aP0, 0, 0);              // next chunk (speculative)
        } else {
#pragma unroll
            for (int p = 0; p < 8; ++p) fsa[p] = aP0[aOff[p]];
            aP0 += aStep;
            __builtin_prefetch(aP0, 0, 0);
        }
        fb0 = *(const float4*)bP0;  fb1 = *(const float4*)bP1;
        bP0 += bStep;  bP1 += bStep;
        __builtin_prefetch(bP0, 0, 0);

        if constexpr (AMODE == 0) {
            *(float4*)&As[wave][q][r4]     = fa0;
            *(float4*)&As[wave][q + 8][r4] = fa1;
        } else {
#pragma unroll
            for (int p = 0; p < 8; ++p) As[wave][p * 2 + half][l16] = fsa[p];
        }
        if constexpr (BMODE == 0) {
            *(float4*)&Bs[wave][q][r4]     = fb0;
            *(float4*)&Bs[wave][q + 8][r4] = fb1;
        } else {   // fb holds B[k=r4..r4+3][n=q(+8)] -> transpose into Bs[k][n]
            Bs[wave][r4+0][q]   = fb0.x;  Bs[wave][r4+1][q]   = fb0.y;
            Bs[wave][r4+2][q]   = fb0.z;  Bs[wave][r4+3][q]   = fb0.w;
            Bs[wave][r4+0][q+8] = fb1.x;  Bs[wave][r4+1][q+8] = fb1.y;
            Bs[wave][r4+2][q+8] = fb1.z;  Bs[wave][r4+3][q+8] = fb1.w;
        }
        __builtin_amdgcn_wave_barrier();   // stores precede the fragment reads

        // ---- 4 x V_WMMA_F32_16X16X4_F32 over this 16-wide k-chunk ----
#pragma unroll
        for (int s = 0; s < 4; ++s) {
            // A 16x4: v0=K{0|2}, v1=K{1|3} across half-waves; M = lane&15
            // B 4x16: v0=rows K{0|2}, v1=rows K{1|3};        N = lane&15
            const int kk = s * 4 + 2 * half;
            v2f av, bv;
            av[0] = As[wave][l16][kk];
            av[1] = As[wave][l16][kk + 1];
            bv[0] = Bs[wave][kk][l16];
            bv[1] = Bs[wave][kk + 1][l16];
            acc = __builtin_amdgcn_wmma_f32_16x16x4_f32(false, av, false, bv,
                                                        (short)0, acc, false, false);
        }
        __builtin_amdgcn_wave_barrier();   // next stores stay after these reads
    }

    const int n_ = tn * 16 + l16;
    const float bval = bias ? bias[n_] : 0.0f;
    const int cbase = blockIdx.z * c_bs + n_ * c_cs;
    const int dbase = blockIdx.z * d_bs + n_ * d_cs;
#pragma unroll
    for (int v = 0; v < 8; ++v) {
        const int m_ = tm * 16 + v + 8 * half;        // C/D layout: VGPR v -> M=v|v+8
        float val = acc[v] + bval;
        if (D)   val += D[dbase + m_ * d_rs];
        if (act) val = gelu_exact(val);
        C[cbase + m_ * c_rs] = val;
    }
}

// ---------------------------------------------------------------------------
// Orchestration. Workspace (floats):
//   basisF 8192 | basisI 8192 | Wfwd2 65536 | Pinv2 65536 |
//   xf/xk 2097152 | z/z2 262144 | h 524288 | byp 16777216   (~79.2 MB)
// ---------------------------------------------------------------------------
extern "C" void kernel_launch(void* const* d_in, const int* in_sizes, int n_in,
                              void* d_out, int out_size, void* d_ws, size_t ws_size,
                              hipStream_t stream) {
    (void)in_sizes; (void)n_in; (void)out_size; (void)ws_size;
    const float* x  = (const float*)d_in[0];
    const float* w0 = (const float*)d_in[1];
    const float* b0 = (const float*)d_in[2];
    const float* w1 = (const float*)d_in[3];
    const float* cw = (const float*)d_in[4];
    const float* cbp= (const float*)d_in[5];
    float* out = (float*)d_out;

    float* ws     = (float*)d_ws;
    float* basisF = ws;
    float* basisI = basisF + 8192;
    float* Wfwd2  = basisI + 8192;
    float* Pinv2  = Wfwd2  + 65536;
    float* xf     = Pinv2  + 65536;     // [r=(b,c,k)][j=2m+ri]  65536x32
    float* z      = xf     + 2097152;   // [b][c*512+l*32+j]     64x4096
    float* h      = z      + 262144;    // [b][n]                64x8192
    float* byp    = h      + 524288;    // [b][o][k][w]          == out layout
    float* xk = xf;                     // reuse: xf consumed before xk written
    float* z2 = z;                      // reuse: z consumed before z2 written

    hipLaunchKernelGGL(init_legendre_k, dim3(1), dim3(128), 0, stream, Wfwd2, Pinv2);
    hipLaunchKernelGGL(init_basis_k,   dim3(32), dim3(256), 0, stream, basisF, basisI);
    hipLaunchKernelGGL(bypass_k,     dim3(2048), dim3(256), 0, stream, x, cw, cbp, byp);

    // 1) forward DFT (m<16): xf = x(65536x256) @ basisF(256x32)
    hipLaunchKernelGGL((gemm_f32_wmma<0,0>), dim3(1024,1,1), dim3(256), 0, stream,
        x, basisF, (const float*)nullptr, (const float*)nullptr, xf,
        RROWS, 32, NLON, 0,
        256,1,0,  32,1,0,  32,1,0,  0,0,0);

    // 2) forward Legendre, batched over j=2m+ri: z[b][c*512+l*32+j]
    hipLaunchKernelGGL((gemm_f32_wmma<1,0>), dim3(4,1,32), dim3(256), 0, stream,
        xf, Wfwd2, (const float*)nullptr, (const float*)nullptr, z,
        BSZ*CIN, 16, NLAT, 0,
        4096,32,1,  16,1,2048,  512,32,1,  0,0,0);

    // 3) MLP layer 1: h = gelu(z(64x4096) @ w0^T(4096x8192) + b0)
    hipLaunchKernelGGL((gemm_f32_wmma<0,1>), dim3(256,1,1), dim3(256), 0, stream,
        z, w0, b0, (const float*)nullptr, h,
        BSZ, HID, INFLT, 1,
        4096,1,0,  1,4096,0,  8192,1,0,  0,0,0);

    // 4) MLP layer 2: z2 = h(64x8192) @ w1^T(8192x4096)
    hipLaunchKernelGGL((gemm_f32_wmma<0,1>), dim3(128,1,1), dim3(256), 0, stream,
        h, w1, (const float*)nullptr, (const float*)nullptr, z2,
        BSZ, INFLT, HID, 0,
        8192,1,0,  1,8192,0,  4096,1,0,  0,0,0);

    // 5) inverse Legendre, batched over j: xk[(r*128+k)*32+j]
    hipLaunchKernelGGL((gemm_f32_wmma<1,0>), dim3(32,1,32), dim3(256), 0, stream,
        z2, Pinv2, (const float*)nullptr, (const float*)nullptr, xk,
        BSZ*COUT, NLAT, 16, 0,
        512,32,1,  128,1,2048,  4096,32,1,  0,0,0);

    // 6) inverse DFT + bypass + gelu: out = gelu(xk(65536x32) @ basisI(32x256) + byp)
    hipLaunchKernelGGL((gemm_f32_wmma<0,0>), dim3(8192,1,1), dim3(256), 0, stream,
        xk, basisI, (const float*)nullptr, byp, out,
        RROWS, NLON, 32, 1,
        32,1,0,  256,1,0,  256,1,0,  256,1,0);
}